// LinearAttention_22943715295778
// MI455X (gfx1250) — compile-verified
//
#include <hip/hip_runtime.h>
#include <hip/hip_bf16.h>
#include <math.h>

typedef float v2f __attribute__((ext_vector_type(2)));
typedef float v8f __attribute__((ext_vector_type(8)));

#define Bsz 4
#define Ntok 8192
#define Mtok 2048
#define Cdim 512
#define Ldim 256
#define Hn 8
#define dL 32
#define dC 64
#define CHUNK 1024
#define SUB 128

__device__ __forceinline__ v8f wmma4(v2f a, v2f b, v8f c) {
    // v_wmma_f32_16x16x4_f32 : D = A(16x4) * B(4x16) + C(16x16), all f32
    return __builtin_amdgcn_wmma_f32_16x16x4_f32(false, a, false, b, (short)0, c, false, false);
}

__global__ void zero_kernel(float* __restrict__ p, int n) {
    int i = blockIdx.x * blockDim.x + threadIdx.x;
    if (i < n) p[i] = 0.0f;
}

// Y = elu(X[rows,Cdim] @ W[Cdim,Ldim] + bias) + 1
// block = 128 (4 waves); wave w does rows [blk.x*128 + w*32, +32), cols [blk.y*64, +64)
template <bool ELU>
__global__ void proj_kernel(const float* __restrict__ X, const float* __restrict__ W,
                            const float* __restrict__ bias, float* __restrict__ Y) {
    const int lane = threadIdx.x & 31;
    const int wave = threadIdx.x >> 5;
    const int row0 = blockIdx.x * 128 + wave * 32;
    const int col0 = blockIdx.y * 64;
    const int m    = lane & 15;
    const int hl   = lane >> 4;      // lane half
    const int koff = hl * 2;

    const float* xrow0 = X + (size_t)(row0 + m) * Cdim + koff;
    const float* xrow1 = X + (size_t)(row0 + 16 + m) * Cdim + koff;
    v8f acc[2][4] = {};
    for (int k0 = 0; k0 < Cdim; k0 += 4) {
        v2f a0, a1;
        a0.x = xrow0[k0];
        a0.y = xrow0[k0 + 1];
        a1.x = xrow1[k0];
        a1.y = xrow1[k0 + 1];
#pragma unroll
        for (int t = 0; t < 4; ++t) {
            const float* wp = W + (size_t)(k0 + koff) * Ldim + col0 + t * 16 + m;
            v2f b;
            b.x = wp[0];
            b.y = wp[Ldim];
            acc[0][t] = wmma4(a0, b, acc[0][t]);
            acc[1][t] = wmma4(a1, b, acc[1][t]);
        }
    }
#pragma unroll
    for (int rt = 0; rt < 2; ++rt) {
#pragma unroll
        for (int t = 0; t < 4; ++t) {
            const int col = col0 + t * 16 + m;
            const float bb = bias[col];
#pragma unroll
            for (int r = 0; r < 8; ++r) {
                const int row = row0 + rt * 16 + r + hl * 8;
                float yv = acc[rt][t][r] + bb;
                if (ELU) yv = (yv > 0.0f) ? (yv + 1.0f) : expf(yv);
                Y[(size_t)row * Ldim + col] = yv;
            }
        }
    }
}

// ksum[b,h,d] = sum_n k[b,n,h*dL+d]
__global__ void ksum_kernel(const float* __restrict__ Kf, float* __restrict__ ksum) {
    const int bh = blockIdx.x;            // 0..B*H-1
    const int b = bh / Hn, h = bh % Hn;
    const int d = threadIdx.x & 31;
    const int part = threadIdx.x >> 5;    // 0..7
    const float* kp = Kf + ((size_t)b * Ntok) * Ldim + h * dL + d;
    float s = 0.0f;
    for (int n = part; n < Ntok; n += 8) s += kp[(size_t)n * Ldim];
    __shared__ float red[8][32];
    red[part][d] = s;
    __syncthreads();
    if (part == 0) {
        float t = 0.0f;
#pragma unroll
        for (int p = 0; p < 8; ++p) t += red[p][d];
        ksum[bh * dL + d] = t;
    }
}

// kv[b,h,c,d] += sum_{n in chunk} V[n,c] * K[n,d], with V computed on the fly.
// grid = B*H*(N/CHUNK), block = 256 (8 waves)
__global__ void kv_kernel(const float* __restrict__ X, const float* __restrict__ Wv,
                          const float* __restrict__ bv, const float* __restrict__ Kf,
                          float* __restrict__ kv) {
    __shared__ float Vlds[SUB][dC + 4];   // 128 x 68 floats = 34 KB

    const int lane = threadIdx.x & 31;
    const int wave = threadIdx.x >> 5;   // 0..7
    const int nchunks = Ntok / CHUNK;    // 8
    const int wg  = blockIdx.x;
    const int b   = wg / (Hn * nchunks);
    const int rem = wg % (Hn * nchunks);
    const int h   = rem / nchunks;
    const int ch  = rem % nchunks;

    const int m    = lane & 15;
    const int hl   = lane >> 4;
    const int koff = hl * 2;

    // stage B mapping: kv output tile (cm: 4 tiles over dC, dn: 2 tiles over dL)
    const int cm = wave & 3;
    const int dn = wave >> 2;

    const size_t tokbase = (size_t)b * Ntok + (size_t)ch * CHUNK;
    float biasv[4];
#pragma unroll
    for (int t = 0; t < 4; ++t) biasv[t] = bv[h * dC + t * 16 + m];

    v8f kvacc = {};

    for (int sc = 0; sc < CHUNK; sc += SUB) {
        // ---- Stage A: V tile (SUB x dC) = X @ Wv[:, h*dC..] into LDS ----
        // wave w computes rows [w*16, +16), all 64 cols (4 tiles, A reused x4)
        {
            const float* xrow = X + (tokbase + sc + wave * 16 + m) * Cdim + koff;
            v8f vacc[4] = {};
            for (int k0 = 0; k0 < Cdim; k0 += 4) {
                v2f a;
                a.x = xrow[k0];
                a.y = xrow[k0 + 1];
#pragma unroll
                for (int t = 0; t < 4; ++t) {
                    const float* wp = Wv + (size_t)(k0 + koff) * Cdim + h * dC + t * 16 + m;
                    v2f bb;
                    bb.x = wp[0];
                    bb.y = wp[Cdim];
                    vacc[t] = wmma4(a, bb, vacc[t]);
                }
            }
#pragma unroll
            for (int t = 0; t < 4; ++t) {
#pragma unroll
                for (int r = 0; r < 8; ++r) {
                    const int row = wave * 16 + r + hl * 8;
                    Vlds[row][t * 16 + m] = vacc[t][r] + biasv[t];
                }
            }
        }
        __syncthreads();
        // ---- Stage B: kvacc += V^T (dC x SUB) * K (SUB x dL) ----
        {
            const float* krow = Kf + (tokbase + sc) * Ldim + h * dL + dn * 16 + m;
#pragma unroll
            for (int n0 = 0; n0 < SUB; n0 += 4) {
                v2f a;
                a.x = Vlds[n0 + koff][cm * 16 + m];
                a.y = Vlds[n0 + koff + 1][cm * 16 + m];
                v2f bb;
                bb.x = krow[(size_t)(n0 + koff) * Ldim];
                bb.y = krow[(size_t)(n0 + koff + 1) * Ldim];
                kvacc = wmma4(a, bb, kvacc);
            }
        }
        __syncthreads();
    }

    float* kvp = kv + (size_t)(b * Hn + h) * dC * dL;
#pragma unroll
    for (int r = 0; r < 8; ++r) {
        const int c = cm * 16 + r + hl * 8;
        const int d = dn * 16 + m;
        atomicAdd(&kvp[c * dL + d], kvacc[r]);
    }
}

// out[b,m,h*dC+c] = z * sum_d q[b,m,h*dL+d] * kv[b,h,c,d];  z = 1/(q . ksum + eps)
// grid = B*H*(M/64), block = 128 (4 waves, each a 16x64 tile)
__global__ void out_kernel(const float* __restrict__ Qf, const float* __restrict__ kv,
                           const float* __restrict__ ksum, float* __restrict__ out) {
    const int lane = threadIdx.x & 31;
    const int wave = threadIdx.x >> 5;
    const int mblocks = Mtok / 64;       // 32
    const int wg  = blockIdx.x;
    const int b   = wg / (Hn * mblocks);
    const int rem = wg % (Hn * mblocks);
    const int h   = rem / mblocks;
    const int mb  = rem % mblocks;
    const int row0 = mb * 64 + wave * 16;

    const int m    = lane & 15;
    const int hl   = lane >> 4;
    const int koff = hl * 2;

    const float* qrow = Qf + ((size_t)b * Mtok + row0 + m) * Ldim + h * dL;
    const float* kvp  = kv + (size_t)(b * Hn + h) * dC * dL;
    const float* ks   = ksum + (b * Hn + h) * dL;

    // normalizer for row (row0 + m)
    float s = 0.0f;
#pragma unroll
    for (int d = 0; d < dL; ++d) s += qrow[d] * ks[d];
    const float zrow = 1.0f / (s + 1e-6f);

    v8f acc[4] = {};
#pragma unroll
    for (int d0 = 0; d0 < dL; d0 += 4) {
        v2f a;
        a.x = qrow[d0 + koff];
        a.y = qrow[d0 + koff + 1];
#pragma unroll
        for (int t = 0; t < 4; ++t) {
            const float* kvq = kvp + (size_t)(t * 16 + m) * dL + d0 + koff;
            v2f bb;
            bb.x = kvq[0];
            bb.y = kvq[1];
            acc[t] = wmma4(a, bb, acc[t]);
        }
    }

#pragma unroll
    for (int r = 0; r < 8; ++r) {
        const int row = row0 + r + hl * 8;
        const float zr = __shfl(zrow, r + hl * 8, 32);
#pragma unroll
        for (int t = 0; t < 4; ++t) {
            const int c = t * 16 + m;
            out[((size_t)b * Mtok + row) * Cdim + h * dC + c] = acc[t][r] * zr;
        }
    }
}

extern "C" void kernel_launch(void* const* d_in, const int* in_sizes, int n_in,
                              void* d_out, int out_size, void* d_ws, size_t ws_size,
                              hipStream_t stream) {
    (void)in_sizes; (void)n_in; (void)out_size; (void)ws_size;
    const float* x  = (const float*)d_in[0];  // input_tokens  [B,N,C]
    const float* qt = (const float*)d_in[1];  // query_tokens  [B,M,C]
    const float* Wq = (const float*)d_in[2];
    const float* bq = (const float*)d_in[3];
    const float* Wk = (const float*)d_in[4];
    const float* bk = (const float*)d_in[5];
    const float* Wv = (const float*)d_in[6];
    const float* bv = (const float*)d_in[7];
    float* out = (float*)d_out;

    float* ws = (float*)d_ws;
    float* qf = ws;                                   // [B*M, L]   8 MB
    float* kf = qf + (size_t)Bsz * Mtok * Ldim;       // [B*N, L]  32 MB
    float* kv = kf + (size_t)Bsz * Ntok * Ldim;       // [B*H, dC, dL] 256 KB
    float* ks = kv + (size_t)Bsz * Hn * dC * dL;      // [B*H, dL]   4 KB

    const int nacc = Bsz * Hn * dC * dL + Bsz * Hn * dL;
    zero_kernel<<<(nacc + 255) / 256, 256, 0, stream>>>(kv, nacc);

    proj_kernel<true><<<dim3(Bsz * Mtok / 128, Ldim / 64), 128, 0, stream>>>(qt, Wq, bq, qf);
    proj_kernel<true><<<dim3(Bsz * Ntok / 128, Ldim / 64), 128, 0, stream>>>(x, Wk, bk, kf);

    ksum_kernel<<<Bsz * Hn, 256, 0, stream>>>(kf, ks);

    kv_kernel<<<Bsz * Hn * (Ntok / CHUNK), 256, 0, stream>>>(x, Wv, bv, kf, kv);

    out_kernel<<<Bsz * Hn * (Mtok / 64), 128, 0, stream>>>(qf, kv, ks, out);
}